// E2EMaskOptWrapper_38268158607497
// MI455X (gfx1250) — compile-verified
//
#include <hip/hip_runtime.h>

typedef __attribute__((ext_vector_type(2))) float v2f;
typedef __attribute__((ext_vector_type(8))) float v8f;

// ---------------------------------------------------------------------------
// Small helper kernels
// ---------------------------------------------------------------------------
__global__ void fill_f32(float* __restrict__ p, float v, int n) {
  int i = blockIdx.x * blockDim.x + threadIdx.x;
  if (i < n) p[i] = v;
}

// w[incident_edge_idx[i]] = edge_gate[incident_group[i]]
__global__ void scatter_gate(float* __restrict__ w,
                             const int* __restrict__ idx,
                             const int* __restrict__ grp,
                             const float* __restrict__ gate, int k) {
  int i = blockIdx.x * blockDim.x + threadIdx.x;
  if (i < k) w[idx[i]] = gate[grp[i]];
}

// deg[dst[e]] += w[e]  (deg pre-filled with 1.0 for the self loop)
__global__ void deg_accum(float* __restrict__ deg,
                          const int* __restrict__ dst,
                          const float* __restrict__ w, int e) {
  int i = blockIdx.x * blockDim.x + threadIdx.x;
  if (i < e) unsafeAtomicAdd(&deg[dst[i]], w[i]);
}

__global__ void rsqrt_inplace(float* __restrict__ d, int n) {
  int i = blockIdx.x * blockDim.x + threadIdx.x;
  if (i < n) d[i] = rsqrtf(d[i]);
}

// gr[k] = x[target*128 + k] * feat_gate[k]   (128 threads)
__global__ void make_gated_row(const float* __restrict__ x,
                               const float* __restrict__ feat_gate,
                               const int* __restrict__ targetPtr,
                               float* __restrict__ gr) {
  int k = threadIdx.x;
  int t = targetPtr[0];
  gr[k] = x[(size_t)t * 128 + k] * feat_gate[k];
}

// ---------------------------------------------------------------------------
// WMMA GEMM:  acc[16x16] = A_tile[16x128] * B[128x128]_tile   (f32, exact)
// One wave -> one 16x16 output tile. blockDim=256 (8 waves) -> 128 output cols.
//
// mode 0:  out[m*ldo+n] = acc + bias[n]          (projection; gated target row)
// mode 1:  out[m*ldo+n] = acc                    (hw for edge gather)
//          agg[m*128+n] = acc*dinv[m]^2 + bias[n] (self-loop term + bias)
// ---------------------------------------------------------------------------
__global__ __launch_bounds__(256)
void gemm_wmma_f32(const float* __restrict__ A, int lda, int Mrows,
                   const float* __restrict__ B,
                   const float* __restrict__ bias,
                   float* __restrict__ out, int ldo,
                   float* __restrict__ agg,
                   const float* __restrict__ dinv,
                   const float* __restrict__ gatedRow,
                   const int* __restrict__ targetPtr,
                   int mode) {
  const int lane    = threadIdx.x & 31;
  const int wave    = threadIdx.x >> 5;
  const int rowBase = blockIdx.x * 16;
  const int colBase = wave * 16;
  const int half    = lane >> 4;        // 0: low half-wave, 1: high half-wave
  const int l15     = lane & 15;
  const int m       = rowBase + l15;    // A row held by this lane
  const int n       = colBase + l15;    // B/C column held by this lane
  const int kb      = half * 2;         // K sub-offset (0 or 2)

  const int tgt  = (mode == 0) ? targetPtr[0] : -1;
  const int mEff = (m < Mrows) ? m : 0;
  const float* __restrict__ Arow =
      (m == tgt) ? gatedRow : (A + (size_t)mEff * (size_t)lda);

  v8f acc = {0.f, 0.f, 0.f, 0.f, 0.f, 0.f, 0.f, 0.f};

#pragma unroll
  for (int k0 = 0; k0 < 128; k0 += 4) {
    const int ka = k0 + kb;
    v2f a, b;
    a.x = Arow[ka];
    a.y = Arow[ka + 1];
    b.x = B[(size_t)ka * 128 + n];
    b.y = B[(size_t)(ka + 1) * 128 + n];
    // D = A(16x4,f32) * B(4x16,f32) + C   -> v_wmma_f32_16x16x4_f32
    acc = __builtin_amdgcn_wmma_f32_16x16x4_f32(
        /*neg_a=*/false, a, /*neg_b=*/false, b,
        /*c_mod=*/(short)0, acc, /*reuse_a=*/false, /*reuse_b=*/false);
  }

  // C/D layout: VGPR r holds M = rowBase + half*8 + r, N = n
  const int mOff = half * 8;
  const float bb = bias[n];
  if (mode == 0) {
#pragma unroll
    for (int r = 0; r < 8; ++r) {
      const int mr = rowBase + mOff + r;
      if (mr < Mrows) out[(size_t)mr * ldo + n] = acc[r] + bb;
    }
  } else {
#pragma unroll
    for (int r = 0; r < 8; ++r) {
      const int mr = rowBase + mOff + r;
      if (mr < Mrows) {
        const float d  = acc[r];
        const float di = dinv[mr];
        out[(size_t)mr * ldo + n] = d;
        agg[(size_t)mr * 128 + n] = d * di * di + bb;  // self-loop + bias
      }
    }
  }
}

// ---------------------------------------------------------------------------
// Edge aggregation: agg[dst] += hw[src] * (w*dinv[src]*dinv[dst])
// One wave per edge; float4 gather; hardware fp32 atomics.
// ---------------------------------------------------------------------------
__global__ __launch_bounds__(256)
void edge_aggregate(const float* __restrict__ hw,
                    float* __restrict__ agg,
                    const int* __restrict__ src,
                    const int* __restrict__ dst,
                    const float* __restrict__ w,
                    const float* __restrict__ dinv, int e) {
  const int edge = blockIdx.x * 8 + (threadIdx.x >> 5);
  if (edge >= e) return;
  const int lane = threadIdx.x & 31;
  const int s = src[edge];
  const int d = dst[edge];
  const float coef = w[edge] * dinv[s] * dinv[d];
  const float4 v = ((const float4*)(hw + (size_t)s * 128))[lane];
  float* ap = agg + (size_t)d * 128 + (size_t)lane * 4;
  unsafeAtomicAdd(ap + 0, v.x * coef);
  unsafeAtomicAdd(ap + 1, v.y * coef);
  unsafeAtomicAdd(ap + 2, v.z * coef);
  unsafeAtomicAdd(ap + 3, v.w * coef);
}

// h[m, colOff + n] = relu(agg[m, n])   written into the [N, 256] concat output
__global__ void relu_store(const float* __restrict__ agg,
                           float* __restrict__ out, int colOff, int total) {
  int i = blockIdx.x * blockDim.x + threadIdx.x;
  if (i >= total) return;
  const int m = i >> 7;
  const int n = i & 127;
  out[(size_t)m * 256 + colOff + n] = fmaxf(agg[i], 0.0f);
}

// ---------------------------------------------------------------------------
// Launch
// ---------------------------------------------------------------------------
extern "C" void kernel_launch(void* const* d_in, const int* in_sizes, int n_in,
                              void* d_out, int out_size, void* d_ws,
                              size_t ws_size, hipStream_t stream) {
  const float* x         = (const float*)d_in[0];
  const float* feat_gate = (const float*)d_in[1];
  const float* edge_gate = (const float*)d_in[2];
  const float* W_proj    = (const float*)d_in[3];
  const float* b_proj    = (const float*)d_in[4];
  const float* W1        = (const float*)d_in[5];
  const float* b1        = (const float*)d_in[6];
  const float* W2        = (const float*)d_in[7];
  const float* b2        = (const float*)d_in[8];
  const int*   ei        = (const int*)d_in[9];    // [2, E]
  const int*   inc_idx   = (const int*)d_in[10];
  const int*   inc_grp   = (const int*)d_in[11];
  const int*   targetPtr = (const int*)d_in[12];

  const int Nn = in_sizes[0] / 128;   // 100000 nodes
  const int Ee = in_sizes[9] / 2;     // 800000 edges
  const int Kk = in_sizes[10];        // 64 incident edges

  const int* srcIdx = ei;
  const int* dstIdx = ei + Ee;

  // Workspace carve-up (floats)
  float* ws    = (float*)d_ws;
  float* w     = ws;                           // [E]   edge weights
  float* dinv  = w + Ee;                       // [N]   deg -> rsqrt(deg)
  float* gr    = dinv + Nn;                    // [128] gated target row
  float* bufH  = gr + 128;                     // [N*128] hw (gather source)
  float* bufA  = bufH + (size_t)Nn * 128;      // [N*128] agg accumulator
  float* bufP  = bufA + (size_t)Nn * 128;      // [N*128] projected features p

  const int rowTiles = (Nn + 15) / 16;

  // 1) edge weights: ones, then incident-edge gates
  fill_f32<<<(Ee + 255) / 256, 256, 0, stream>>>(w, 1.0f, Ee);
  scatter_gate<<<(Kk + 63) / 64, 64, 0, stream>>>(w, inc_idx, inc_grp,
                                                  edge_gate, Kk);
  // 2) degree (self loop = 1) -> dinv = rsqrt(deg)
  fill_f32<<<(Nn + 255) / 256, 256, 0, stream>>>(dinv, 1.0f, Nn);
  deg_accum<<<(Ee + 255) / 256, 256, 0, stream>>>(dinv, dstIdx, w, Ee);
  rsqrt_inplace<<<(Nn + 255) / 256, 256, 0, stream>>>(dinv, Nn);

  // 3) gated target row, then projection p = x_m @ W_proj + b_proj
  make_gated_row<<<1, 128, 0, stream>>>(x, feat_gate, targetPtr, gr);
  gemm_wmma_f32<<<rowTiles, 256, 0, stream>>>(x, 128, Nn, W_proj, b_proj,
                                              bufP, 128, nullptr, nullptr, gr,
                                              targetPtr, /*mode=*/0);

  // 4) layer 1: hw1 = p @ W1 ; agg = hw1/deg + b1 ; scatter ; h1 = relu
  gemm_wmma_f32<<<rowTiles, 256, 0, stream>>>(bufP, 128, Nn, W1, b1, bufH, 128,
                                              bufA, dinv, gr, targetPtr,
                                              /*mode=*/1);
  edge_aggregate<<<(Ee + 7) / 8, 256, 0, stream>>>(bufH, bufA, srcIdx, dstIdx,
                                                   w, dinv, Ee);
  relu_store<<<((Nn * 128) + 255) / 256, 256, 0, stream>>>(
      bufA, (float*)d_out, /*colOff=*/0, Nn * 128);

  // 5) layer 2: hw2 = h1 @ W2 (read h1 from d_out, lda=256) ; same pipeline
  gemm_wmma_f32<<<rowTiles, 256, 0, stream>>>((const float*)d_out, 256, Nn, W2,
                                              b2, bufH, 128, bufA, dinv, gr,
                                              targetPtr, /*mode=*/1);
  edge_aggregate<<<(Ee + 7) / 8, 256, 0, stream>>>(bufH, bufA, srcIdx, dstIdx,
                                                   w, dinv, Ee);
  relu_store<<<((Nn * 128) + 255) / 256, 256, 0, stream>>>(
      bufA, (float*)d_out, /*colOff=*/128, Nn * 128);
}